// UltraHighPrecisionNKATHamiltonian_38585986187668
// MI455X (gfx1250) — compile-verified
//
#include <hip/hip_runtime.h>
#include <hip/hip_bf16.h>
#include <math.h>

// CDNA5 / gfx1250: wave32, WMMA f32 16x16x4
typedef __attribute__((ext_vector_type(2))) float v2f;
typedef __attribute__((ext_vector_type(8))) float v8f;

#define DIMN   300     // Hamiltonian dimension
#define CDIM   128     // coupled sub-block: all off-diagonals live at idx < 114
#define NKAP   40
#define THETA  1e-19f  // 1e-20 * 10
#define KAPPA  5e-12f  // 1e-12 * 5
#define SORTN  512
#define SVTHR  1e-12f
#define POSTHR 1e-20f

__device__ __forceinline__ bool is_prime_v(int v) {
  // primes[:30] = 2..113 as two 64-bit masks
  const unsigned long long lo =
      (1ull<<2)|(1ull<<3)|(1ull<<5)|(1ull<<7)|(1ull<<11)|(1ull<<13)|(1ull<<17)|
      (1ull<<19)|(1ull<<23)|(1ull<<29)|(1ull<<31)|(1ull<<37)|(1ull<<41)|(1ull<<43)|
      (1ull<<47)|(1ull<<53)|(1ull<<59)|(1ull<<61);
  const unsigned long long hi =
      (1ull<<(67-64))|(1ull<<(71-64))|(1ull<<(73-64))|(1ull<<(79-64))|(1ull<<(83-64))|
      (1ull<<(89-64))|(1ull<<(97-64))|(1ull<<(101-64))|(1ull<<(103-64))|(1ull<<(107-64))|
      (1ull<<(109-64))|(1ull<<(113-64));
  if ((unsigned)v >= 128u) return false;
  unsigned long long m = (v < 64) ? (lo >> v) : (hi >> (v - 64));
  return (m & 1ull) != 0ull;
}

__device__ __forceinline__ float kap_c(int i) {  // Minkowski-damped kappa coupling
  float nn = (float)(i + 1);
  float mk = rsqrtf(1.f + (nn * KAPPA) * (nn * KAPPA));
  return KAPPA * nn * logf(nn + 1.f) * mk;
}

__device__ __forceinline__ float prime_corr(int p, float cs) {
  return THETA * logf((float)p) * cs;
}

__global__ __launch_bounds__(256)
void nkat_eigs_kernel(const float* __restrict__ s_real,
                      const float* __restrict__ s_imag,
                      float* __restrict__ out, int n_eig) {
  __shared__ float dv[DIMN];       // Hh diagonal (all 300 modes)
  __shared__ float sd_s[CDIM];     // Hh[i][i+1].re  (kappa tridiag, symmetric)
  __shared__ float pi_s[CDIM];     // Hh[i][i+1].im  (prime skew, anti-symmetric)
  __shared__ float sq[CDIM];       // diag(Hh^2) on coupled block
  __shared__ float csh[8 * 256];   // per-wave 16x16 C-tile spill
  __shared__ float sm[SORTN];      // sort buffer

  const int tid = threadIdx.x;
  const int b   = blockIdx.x;
  __builtin_prefetch(&s_imag[b], 0, 0);     // global_prefetch_b8
  const float sr    = s_real[b];
  const float si    = s_imag[b];
  const float gamma = fabsf(si);
  const float cs    = 1.f / (1.f + gamma * 0.01f);
  const float reg   = 1e-15f * (1.f + gamma * 1e-4f);

  // ---- build diagonal + the two sparse coupling bands once ----
  for (int n = tid; n < DIMN; n += 256) {
    float fn = (float)(n + 1);
    float ln = logf(fn);
    float d  = expf(-sr * ln) * cosf(si * ln);     // Re(n^{-s})
    if (n < NKAP)          d += kap_c(n) * 0.01f;  // kappa diag
    if (is_prime_v(n + 1)) d += prime_corr(n + 1, cs) * 0.1f;
    dv[n] = d + reg;
  }
  for (int i = tid; i < CDIM; i += 256) {
    sd_s[i] = (i < NKAP) ? kap_c(i) * 0.05f : 0.f;                    // Hh[i][i+1].re
    pi_s[i] = is_prime_v(i + 1) ? 0.5f * prime_corr(i + 1, cs) : 0.f; // Hh[i][i+1].im
  }
  __syncthreads();

  // ---- WMMA stage: wave w computes diagonal tile w of Hh^2 over the 128x128
  // coupled block. C_ij = sum_k (Hr[i][k]Hr[j][k] + Hi[i][k]Hi[j][k]).
  // Hh is tridiagonal-sparse: tile w's rows only touch columns in the band
  // [16w-1, 16w+16], so the K-chain is restricted to the 32-column window
  // [kbase, kbase+32) with kbase = max(0, 16w-8) (4-aligned): 8 unrolled steps,
  // 16 WMMAs per wave instead of 64. Out-of-band fragments compare to zero in
  // registers (no memory is touched), so the wave-7 window past column 127 is
  // benign. Fragment selects are branch-free v_cndmask -> EXEC stays all-ones.
  const int wave  = tid >> 5;
  const int lane  = tid & 31;
  const int r     = 16 * wave + (lane & 15);
  const int khalf = (lane >> 4) * 2;
  const int kbase = (wave == 0) ? 0 : (16 * wave - 8);
  const float dr  = dv[r];
  const float sdr = sd_s[r];                      // couples (r, r+1)
  const float sdl = (r > 0) ? sd_s[r - 1] : 0.f;  // couples (r-1, r)
  const float pir = pi_s[r];
  const float pil = (r > 0) ? -pi_s[r - 1] : 0.f; // conj for (r, r-1)

  v8f acc = {};
#pragma unroll
  for (int kk = 0; kk < 32; kk += 4) {
    const int ka = kbase + kk + khalf;
    const int kb = ka + 1;
    float arx = (ka == r) ? dr : (ka == r + 1) ? sdr : (ka == r - 1) ? sdl : 0.f;
    float ary = (kb == r) ? dr : (kb == r + 1) ? sdr : (kb == r - 1) ? sdl : 0.f;
    float aix = (ka == r + 1) ? pir : (ka == r - 1) ? pil : 0.f;
    float aiy = (kb == r + 1) ? pir : (kb == r - 1) ? pil : 0.f;
    v2f ar; ar.x = arx; ar.y = ary;
    v2f ai; ai.x = aix; ai.y = aiy;
    // A and B fragments coincide for P*P^T under the 16x4 f32 layout
    acc = __builtin_amdgcn_wmma_f32_16x16x4_f32(false, ar, false, ar,
                                                (short)0, acc, false, false);
    acc = __builtin_amdgcn_wmma_f32_16x16x4_f32(false, ai, false, ai,
                                                (short)0, acc, false, false);
  }
  {  // spill C tile (VGPR v: lanes0-15 -> M=v, lanes16-31 -> M=v+8; N=lane%16)
    float* ct = &csh[wave * 256];
    const int nn = lane & 15;
#pragma unroll
    for (int v = 0; v < 8; ++v) {
      int m = v + ((lane < 16) ? 0 : 8);
      ct[m * 16 + nn] = acc[v];
    }
  }
  __syncthreads();
  for (int i = tid; i < CDIM; i += 256)
    sq[i] = csh[(i >> 4) * 256 + (i & 15) * 17];  // local (m,m) of tile i>>4
  __syncthreads();

  // ---- eigenvalue estimates + SVD sigma-floor + positivity mask ----
  for (int n = tid; n < SORTN; n += 256) {
    float lam;
    if (n < CDIM) {                       // second-order magnitude refinement
      float mag = sqrtf(sq[n]);
      lam = (dv[n] < 0.f) ? -mag : mag;
    } else if (n < DIMN) {
      lam = dv[n];                        // purely diagonal modes
    } else {
      lam = -INFINITY;                    // sort padding
    }
    if (n < DIMN) {
      if (lam > 0.f)      lam = fmaxf(lam,  SVTHR);   // sigma floor
      else if (lam < 0.f) lam = fminf(lam, -SVTHR);
      lam = (lam > POSTHR) ? lam : -INFINITY;         // positive mask
    }
    sm[n] = lam;
  }
  __syncthreads();

  // ---- bitonic sort, descending, 512 elems / 256 threads ----
  for (int k = 2; k <= SORTN; k <<= 1) {
    for (int j = k >> 1; j > 0; j >>= 1) {
      for (int i = tid; i < SORTN; i += 256) {
        int ixj = i ^ j;
        if (ixj > i) {
          float a = sm[i], c = sm[ixj];
          bool desc = ((i & k) == 0);
          if (desc ? (a < c) : (a > c)) { sm[i] = c; sm[ixj] = a; }
        }
      }
      __syncthreads();
    }
  }

  for (int j = tid; j < n_eig; j += 256) {
    float v = sm[j];
    out[(size_t)b * n_eig + j] = isfinite(v) ? v : 0.f;
  }
}

extern "C" void kernel_launch(void* const* d_in, const int* in_sizes, int n_in,
                              void* d_out, int out_size, void* d_ws, size_t ws_size,
                              hipStream_t stream) {
  (void)n_in; (void)d_ws; (void)ws_size;
  const float* s_real = (const float*)d_in[0];
  const float* s_imag = (const float*)d_in[1];
  // d_in[2] = n_eigenvalues (device int scalar); derive deterministically instead:
  int batch = in_sizes[0];
  if (batch <= 0) return;
  int n_eig = out_size / batch;
  if (n_eig <= 0) return;
  nkat_eigs_kernel<<<batch, 256, 0, stream>>>(s_real, s_imag, (float*)d_out, n_eig);
}